// SpVecsUnet_18348100288870
// MI455X (gfx1250) — compile-verified
//
#include <hip/hip_runtime.h>
#include <cstdint>

// Problem constants (fixed in the reference file)
#define NPIX  (1u << 20)   // H*W = 1024*1024
#define NSEG  4096         // S
#define NCH   64           // C
#define CGRP  16           // channels accumulated per block (LDS table = 16*4096*4 = 256KB)
#define ACC_THREADS 512    // 16 wave32 waves
#define PIX_BLOCKS 32      // pixel chunks along grid.x

#define CPY_THREADS 256
#define CPY_BLOCKS  8192
#define CPY_BATCH   4      // async chunks in flight per thread

#ifndef __has_builtin
#define __has_builtin(x) 0
#endif

#if __has_builtin(__builtin_amdgcn_global_load_async_to_lds_b128) && \
    __has_builtin(__builtin_amdgcn_global_store_async_from_lds_b128) && \
    __has_builtin(__builtin_amdgcn_s_wait_asynccnt)
#define HAVE_ASYNC_LDS 1
#else
#define HAVE_ASYNC_LDS 0
#endif

// b128 async payload type: 4 x i32 vector, in global (AS1) and LDS (AS3) spaces
typedef int v4i __attribute__((vector_size(16)));
typedef v4i __attribute__((address_space(1))) as1_v4i;
typedef v4i __attribute__((address_space(3))) as3_v4i;

// ---------------- zero the accumulator workspace ----------------
__global__ void zero_ws_kernel(float* __restrict__ ws, int n) {
    int i = blockIdx.x * blockDim.x + threadIdx.x;
    if (i < n) ws[i] = 0.0f;
}

// ---------------- LDS-privatized segment accumulate ----------------
// grid = (PIX_BLOCKS, NCH/CGRP); block = ACC_THREADS; dynLDS = (CGRP+1)*NSEG*4
__global__ __launch_bounds__(ACC_THREADS)
void accumulate_kernel(const float* __restrict__ features,
                       const int*   __restrict__ pix_idx,
                       const int*   __restrict__ seg_ids,
                       float*       __restrict__ sums,     // [NSEG][NCH]
                       float*       __restrict__ counts) { // [NSEG]
    extern __shared__ float smem[];            // lsum[CGRP][NSEG] then lcnt[NSEG]
    float* lsum = smem;
    float* lcnt = smem + CGRP * NSEG;
    const int tid = threadIdx.x;
    const int cg  = blockIdx.y;
    const int c0  = cg * CGRP;

    for (int i = tid; i < (CGRP + 1) * NSEG; i += ACC_THREADS) smem[i] = 0.0f;
    __syncthreads();

    const int per   = NPIX / PIX_BLOCKS;
    const int start = blockIdx.x * per;
    const int end   = start + per;
    for (int p = start + tid; p < end; p += ACC_THREADS) {
        const unsigned pix = (unsigned)pix_idx[p];   // gather index (arange in practice)
        const int      seg = seg_ids[p];
        if (cg == 0) atomicAdd(&lcnt[seg], 1.0f);    // ds_add_f32
        #pragma unroll
        for (int c = 0; c < CGRP; ++c) {
            // coalesced across lanes for each c; LDS bank = seg & 63 (uniform random)
            float v = features[(size_t)(c0 + c) * NPIX + pix];
            atomicAdd(&lsum[c * NSEG + seg], v);     // ds_add_f32
        }
    }
    __syncthreads();

    // flush block-private table to global with native fp32 atomics
    for (int i = tid; i < CGRP * NSEG; i += ACC_THREADS) {
        const int c   = i >> 12;          // i / NSEG
        const int seg = i & (NSEG - 1);
        unsafeAtomicAdd(&sums[(size_t)seg * NCH + c0 + c], lsum[i]);
    }
    if (cg == 0) {
        for (int i = tid; i < NSEG; i += ACC_THREADS)
            unsafeAtomicAdd(&counts[i], lcnt[i]);
    }
}

// ---------------- divide by max(count, 1) ----------------
__global__ __launch_bounds__(256)
void finalize_kernel(const float* __restrict__ sums,
                     const float* __restrict__ counts,
                     float* __restrict__ out) {
    int i = blockIdx.x * blockDim.x + threadIdx.x;   // < NSEG*NCH
    float cnt = counts[i >> 6];                      // i / NCH
    out[i] = sums[i] / fmaxf(cnt, 1.0f);
}

// ---------------- feature passthrough copy (batched async-LDS path) ----------------
// n16 (2^24) divides exactly: each thread moves 8 chunks = 2 groups of CPY_BATCH.
__global__ __launch_bounds__(CPY_THREADS)
void copy_features_kernel(const float4* __restrict__ src,
                          float4* __restrict__ dst, unsigned n16) {
    const unsigned stride = gridDim.x * blockDim.x;
    const unsigned i0 = blockIdx.x * blockDim.x + threadIdx.x;
#if HAVE_ASYNC_LDS
    __shared__ float4 stage[CPY_THREADS * CPY_BATCH];
    as3_v4i* lp[CPY_BATCH];
    #pragma unroll
    for (int k = 0; k < CPY_BATCH; ++k)
        lp[k] = (as3_v4i*)(unsigned)(uintptr_t)&stage[threadIdx.x * CPY_BATCH + k];

    // groups of CPY_BATCH grid-strided chunks; each load/store is lane-coalesced
    for (unsigned i = i0; i + (CPY_BATCH - 1) * stride < n16; i += CPY_BATCH * stride) {
        #pragma unroll
        for (int k = 0; k < CPY_BATCH; ++k)
            __builtin_amdgcn_global_load_async_to_lds_b128(
                (as1_v4i*)(uintptr_t)(src + i + k * stride), lp[k], 0, 0);
        __builtin_amdgcn_s_wait_asynccnt(0);     // all 4 loads landed in LDS
        #pragma unroll
        for (int k = 0; k < CPY_BATCH; ++k)
            __builtin_amdgcn_global_store_async_from_lds_b128(
                (as1_v4i*)(uintptr_t)(dst + i + k * stride), lp[k], 0, 0);
        __builtin_amdgcn_s_wait_asynccnt(0);     // stores read LDS before slot reuse
    }
#else
    for (unsigned i = i0; i < n16; i += stride)
        dst[i] = src[i];
#endif
}

extern "C" void kernel_launch(void* const* d_in, const int* in_sizes, int n_in,
                              void* d_out, int out_size, void* d_ws, size_t ws_size,
                              hipStream_t stream) {
    const float* features = (const float*)d_in[0];   // [NCH, NPIX]
    const int*   pix_idx  = (const int*)d_in[1];     // [NPIX]
    const int*   seg_ids  = (const int*)d_in[2];     // [NPIX]
    (void)in_sizes; (void)n_in; (void)out_size; (void)ws_size;

    float* out_vecs = (float*)d_out;                         // [NSEG*NCH]
    float* out_feat = out_vecs + (size_t)NSEG * NCH;         // [NCH*NPIX]
    float* sums     = (float*)d_ws;                          // [NSEG*NCH]
    float* counts   = sums + (size_t)NSEG * NCH;             // [NSEG]

    // 1) zero accumulators (must happen every call: graph replays re-accumulate)
    {
        const int n = NSEG * NCH + NSEG;
        zero_ws_kernel<<<(n + 255) / 256, 256, 0, stream>>>(sums, n);
    }
    // 2) LDS-privatized scatter-add
    {
        dim3 grid(PIX_BLOCKS, NCH / CGRP);
        size_t shmem = (size_t)(CGRP + 1) * NSEG * sizeof(float);  // 272 KB (CDNA5 WGP LDS)
        accumulate_kernel<<<grid, ACC_THREADS, shmem, stream>>>(
            features, pix_idx, seg_ids, sums, counts);
    }
    // 3) means
    finalize_kernel<<<(NSEG * NCH) / 256, 256, 0, stream>>>(sums, counts, out_vecs);
    // 4) passthrough copy of features (second tuple element)
    {
        const unsigned n16 = (unsigned)((size_t)NCH * NPIX / 4);   // 16M float4 chunks
        copy_features_kernel<<<CPY_BLOCKS, CPY_THREADS, 0, stream>>>(
            (const float4*)features, (float4*)out_feat, n16);
    }
}